// TTLM_57440892617008
// MI455X (gfx1250) — compile-verified
//
#include <hip/hip_runtime.h>

// ---------------------------------------------------------------------------
// Linear recurrence h_t = A h_{t-1} + Bm x_t, B=16, L=2048, H=N=512, fp32.
// Chunked scan, T=32, C=64 chunks, plus a TWO-LEVEL inter-chunk scan
// (8 groups x 8) so no single workgroup ever runs more than ~8 dependent
// 512x512x16 steps. Serial depth: 32 (pass1) + 22 (scan) + 32 (pass3) ~ 86
// dependent WMMA-matmul steps vs 2048 naive. All matmuls use
// V_WMMA_F32_16X16X4_F32 (fp32 preserves reference precision over the long
// recurrence; A streams from the 192MB L2 where it stays hot).
// Workspace use: ~72 MB.
// ---------------------------------------------------------------------------

typedef __attribute__((ext_vector_type(2))) float v2f;
typedef __attribute__((ext_vector_type(4))) float v4f;
typedef __attribute__((ext_vector_type(8))) float v8f;

#define B_SZ 16
#define LSEQ 2048
#define HDIM 512
#define NDIM 512
#define TCH  32            // chunk length
#define CCH  (LSEQ / TCH)  // 64 chunks
#define GRP  8             // groups in two-level scan
#define GSZ  (CCH / GRP)   // 8 chunks per group
#define NT   (NDIM / 16)   // 32 row tiles of the (N,B) state
#define HPAD 17            // LDS row stride (floats) to break bank conflicts

// D = A(16x4) * B(4x16) + C, fp32 WMMA
__device__ __forceinline__ v8f wmma4(v2f a, v2f b, v8f c) {
  return __builtin_amdgcn_wmma_f32_16x16x4_f32(
      /*neg_a=*/false, a, /*neg_b=*/false, b,
      /*c_mod=*/(short)0, c, /*reuse_a=*/false, /*reuse_b=*/false);
}

// A-fragment (16x4) from a row-major matrix with leading dim ld.
// Lanes 0-15 -> M=lane, K={k,k+1}; lanes 16-31 -> M=lane-16, K={k+2,k+3}
__device__ __forceinline__ v2f load_afrag(const float* __restrict__ M, int ld,
                                          int row0, int k, int lane) {
  const float* p = M + (size_t)(row0 + (lane & 15)) * ld + k + ((lane >> 4) << 1);
  return v2f{p[0], p[1]};
}

// B-fragment (4x16) from LDS state hlds[k*HPAD + b] (K-major, padded).
__device__ __forceinline__ v2f load_hfrag(const float* __restrict__ hlds,
                                          int k, int lane) {
  int b  = lane & 15;
  int kk = k + ((lane >> 4) << 1);
  return v2f{hlds[kk * HPAD + b], hlds[(kk + 1) * HPAD + b]};
}

// ---------------------------------------------------------------------------
// Pass 0: P[l, tile] = Bm[tile rows] @ x_l^T   (WMMA fragment layout:
// float P[L][NT][32 lanes][8])
// ---------------------------------------------------------------------------
__global__ void proj_kernel(const float* __restrict__ x,   // (B, L, H)
                            const float* __restrict__ Bm,  // (N, H)
                            float* __restrict__ P) {
  int wid  = (int)((blockIdx.x * blockDim.x + threadIdx.x) >> 5);
  int lane = (int)(threadIdx.x & 31);
  int l    = wid >> 5;   // / NT
  int tile = wid & 31;   // % NT
  int row0 = tile * 16;
  int b     = lane & 15;
  int khalf = (lane >> 4) << 1;
  const float* xb = x + ((size_t)b * LSEQ + l) * HDIM;  // B[k][b] = x[b][l][k]
  v8f acc = {};
  for (int k = 0; k < HDIM; k += 4) {
    v2f a = load_afrag(Bm, HDIM, row0, k, lane);
    const float* q = xb + k + khalf;
    v2f bb = v2f{q[0], q[1]};
    acc = wmma4(a, bb, acc);
  }
  *(v8f*)(P + (((size_t)l * NT + tile) * 32 + lane) * 8) = acc;
}

// ---------------------------------------------------------------------------
// 512x512 fp32 matmul (for A^2 ... A^256 by repeated squaring)
// ---------------------------------------------------------------------------
__global__ void mm512_kernel(const float* __restrict__ M1,
                             const float* __restrict__ M2,
                             float* __restrict__ out) {
  int wid  = (int)((blockIdx.x * blockDim.x + threadIdx.x) >> 5);
  int lane = (int)(threadIdx.x & 31);
  int tr = wid >> 5, tc = wid & 31;
  int row0 = tr * 16;
  int col  = tc * 16 + (lane & 15);
  int khalf = (lane >> 4) << 1;
  v8f acc = {};
  for (int k = 0; k < NDIM; k += 4) {
    v2f a = load_afrag(M1, NDIM, row0, k, lane);
    int kk = k + khalf;
    v2f bb = v2f{M2[(size_t)kk * NDIM + col], M2[(size_t)(kk + 1) * NDIM + col]};
    acc = wmma4(a, bb, acc);
  }
  int r0 = row0 + ((lane >> 4) << 3);
#pragma unroll
  for (int v = 0; v < 8; ++v) out[(size_t)(r0 + v) * NDIM + col] = acc[v];
}

// ---------------------------------------------------------------------------
// Passes 1 & 3: per-chunk serial recurrence over P (16 waves, 2 row tiles
// each; h (512x16) in padded LDS; A streamed from L2).
// mode 0: zero init, emit chunk-end state E[c].
// mode 1: init from Hstart[c], emit outputs (B,L,N) and h_final (N,B).
// ---------------------------------------------------------------------------
__global__ void chunk_kernel(const float* __restrict__ A,       // (N,N)
                             const float* __restrict__ P,       // frag layout
                             const float* __restrict__ Hstart,  // frag layout (mode 1)
                             float* __restrict__ E,             // frag layout (mode 0)
                             float* __restrict__ out,           // (B,L,N)+(N,B) (mode 1)
                             int mode) {
  __shared__ float hlds[NDIM * HPAD];
  int c    = (int)blockIdx.x;
  int wave = (int)(threadIdx.x >> 5);
  int lane = (int)(threadIdx.x & 31);
  int b    = lane & 15;
  int rsel = lane >> 4;

#pragma unroll
  for (int t = 0; t < 2; ++t) {
    int tile = wave * 2 + t;
    int n0   = tile * 16 + rsel * 8;
    v8f s = {};
    if (mode == 1)
      s = *(const v8f*)(Hstart + (((size_t)c * NT + tile) * 32 + lane) * 8);
#pragma unroll
    for (int v = 0; v < 8; ++v) hlds[(n0 + v) * HPAD + b] = s[v];
  }
  __syncthreads();

  for (int tau = 0; tau < TCH; ++tau) {
    int l = c * TCH + tau;
    v8f acc[2];
#pragma unroll
    for (int t = 0; t < 2; ++t) {
      int tile = wave * 2 + t;
      acc[t] = *(const v8f*)(P + (((size_t)l * NT + tile) * 32 + lane) * 8);
      int row0 = tile * 16;
      for (int k = 0; k < NDIM; k += 4) {
        v2f a  = load_afrag(A, NDIM, row0, k, lane);
        v2f hb = load_hfrag(hlds, k, lane);
        acc[t] = wmma4(a, hb, acc[t]);
      }
    }
    __syncthreads();  // all reads of old h done
#pragma unroll
    for (int t = 0; t < 2; ++t) {
      int tile = wave * 2 + t;
      int n0   = tile * 16 + rsel * 8;
#pragma unroll
      for (int v = 0; v < 8; ++v) hlds[(n0 + v) * HPAD + b] = acc[t][v];
      if (mode == 1) {
        // lane's 8 accumulators are 8 consecutive n values -> contiguous store
        float* op = out + (((size_t)b * LSEQ + l) * NDIM + n0);
        *(v4f*)op       = v4f{acc[t][0], acc[t][1], acc[t][2], acc[t][3]};
        *(v4f*)(op + 4) = v4f{acc[t][4], acc[t][5], acc[t][6], acc[t][7]};
      }
      if (mode == 0 && tau == TCH - 1)
        *(v8f*)(E + (((size_t)c * NT + tile) * 32 + lane) * 8) = acc[t];
      if (mode == 1 && tau == TCH - 1 && c == CCH - 1) {
        float* hf = out + (size_t)B_SZ * LSEQ * NDIM;  // h_final (N,B)
#pragma unroll
        for (int v = 0; v < 8; ++v) hf[(size_t)(n0 + v) * B_SZ + b] = acc[t][v];
      }
    }
    __syncthreads();
  }
}

// ---------------------------------------------------------------------------
// Generic short scan:  h := (Cin ? Cin[blockIdx] : 0);
//   optionally store h, then repeat `steps` times: h = M h + Add[base + j],
//   storing each state (Sout) and/or the final state (Send).
// Used for all three phases of the two-level inter-chunk scan.
// ---------------------------------------------------------------------------
__global__ void scan2_kernel(const float* __restrict__ M,    // (N,N)
                             const float* __restrict__ Add,  // frag layout
                             const float* __restrict__ Cin,  // frag @ blockIdx, or null
                             float* __restrict__ Sout,       // frag, steps+1 states @ blockIdx*(steps+1)
                             float* __restrict__ Send,       // frag, final state @ blockIdx
                             int steps, int addStride) {
  __shared__ float hlds[NDIM * HPAD];
  int g    = (int)blockIdx.x;
  int wave = (int)(threadIdx.x >> 5);
  int lane = (int)(threadIdx.x & 31);
  int b    = lane & 15;
  int rsel = lane >> 4;

#pragma unroll
  for (int t = 0; t < 2; ++t) {
    int tile = wave * 2 + t;
    int n0   = tile * 16 + rsel * 8;
    v8f s = {};
    if (Cin) s = *(const v8f*)(Cin + (((size_t)g * NT + tile) * 32 + lane) * 8);
#pragma unroll
    for (int v = 0; v < 8; ++v) hlds[(n0 + v) * HPAD + b] = s[v];
    if (Sout)
      *(v8f*)(Sout + (((size_t)(g * (steps + 1)) * NT + tile) * 32 + lane) * 8) = s;
  }
  __syncthreads();

  for (int j = 0; j < steps; ++j) {
    v8f acc[2];
#pragma unroll
    for (int t = 0; t < 2; ++t) {
      int tile = wave * 2 + t;
      acc[t] = *(const v8f*)(Add + (((size_t)(g * addStride + j) * NT + tile) * 32 + lane) * 8);
      int row0 = tile * 16;
      for (int k = 0; k < NDIM; k += 4) {
        v2f a  = load_afrag(M, NDIM, row0, k, lane);
        v2f hb = load_hfrag(hlds, k, lane);
        acc[t] = wmma4(a, hb, acc[t]);
      }
    }
    __syncthreads();
#pragma unroll
    for (int t = 0; t < 2; ++t) {
      int tile = wave * 2 + t;
      int n0   = tile * 16 + rsel * 8;
#pragma unroll
      for (int v = 0; v < 8; ++v) hlds[(n0 + v) * HPAD + b] = acc[t][v];
      if (Sout)
        *(v8f*)(Sout + (((size_t)(g * (steps + 1) + j + 1) * NT + tile) * 32 + lane) * 8) = acc[t];
      if (Send && j == steps - 1)
        *(v8f*)(Send + (((size_t)g * NT + tile) * 32 + lane) * 8) = acc[t];
    }
    __syncthreads();
  }
}

// ---------------------------------------------------------------------------
extern "C" void kernel_launch(void* const* d_in, const int* in_sizes, int n_in,
                              void* d_out, int out_size, void* d_ws, size_t ws_size,
                              hipStream_t stream) {
  const float* x  = (const float*)d_in[0];  // (B, L, H)
  const float* A  = (const float*)d_in[1];  // (N, N)
  const float* Bm = (const float*)d_in[2];  // (N, H)
  float* out = (float*)d_out;               // (B,L,N) ++ (N,B)

  // Workspace layout (floats): P | W0 | W1 | W2 | E | Hstart | Eg | G  (~72 MB)
  float* P  = (float*)d_ws;
  float* W0 = P + (size_t)LSEQ * NDIM * B_SZ;           // holds A^32 at the end
  float* W1 = W0 + (size_t)NDIM * NDIM;                 // holds A^256 at the end
  float* W2 = W1 + (size_t)NDIM * NDIM;
  float* E  = W2 + (size_t)NDIM * NDIM;                 // chunk-end states, frag
  float* Hs = E + (size_t)CCH * NDIM * B_SZ;            // chunk-start states, frag
  float* Eg = Hs + (size_t)CCH * NDIM * B_SZ;           // group-end states, frag
  float* G  = Eg + (size_t)GRP * NDIM * B_SZ;           // group-start states, frag

  // Pass 0: input projection (2048*32 wave-tiles, 4 waves/block)
  proj_kernel<<<(LSEQ * NT) / 4, 128, 0, stream>>>(x, Bm, P);

  // Repeated squaring: A2->W0, A4->W1, A8->W2, A16->W1, A32->W0,
  //                    A64->W1, A128->W2, A256->W1
  mm512_kernel<<<256, 128, 0, stream>>>(A, A, W0);
  mm512_kernel<<<256, 128, 0, stream>>>(W0, W0, W1);
  mm512_kernel<<<256, 128, 0, stream>>>(W1, W1, W2);
  mm512_kernel<<<256, 128, 0, stream>>>(W2, W2, W1);
  mm512_kernel<<<256, 128, 0, stream>>>(W1, W1, W0);   // W0 = A^32
  mm512_kernel<<<256, 128, 0, stream>>>(W0, W0, W1);
  mm512_kernel<<<256, 128, 0, stream>>>(W1, W1, W2);
  mm512_kernel<<<256, 128, 0, stream>>>(W2, W2, W1);   // W1 = A^256
  const float* A32  = W0;
  const float* A256 = W1;

  // Pass 1: parallel chunk-local scans (zero init) -> E[c]
  chunk_kernel<<<CCH, 512, 0, stream>>>(A, P, nullptr, E, nullptr, 0);

  // Two-level inter-chunk scan:
  // S1: per-group local scans (8 steps, zero init) -> group sums Eg[g]
  scan2_kernel<<<GRP, 512, 0, stream>>>(A32, E, nullptr, nullptr, Eg, GSZ, GSZ);
  // S2: top-level scan of 8 group carries with A^256 -> G[0..GRP-1]
  scan2_kernel<<<1, 512, 0, stream>>>(A256, Eg, nullptr, G, nullptr, GRP - 1, 0);
  // S3: replay each group from its true start G[g] -> Hstart[8g..8g+7]
  scan2_kernel<<<GRP, 512, 0, stream>>>(A32, E, G, Hs, nullptr, GSZ - 1, GSZ);

  // Pass 3: replay chunks from corrected starts -> outputs + h_final
  chunk_kernel<<<CCH, 512, 0, stream>>>(A, P, Hs, nullptr, out, 1);
}